// AML_TGNN_80882824118638
// MI455X (gfx1250) — compile-verified
//
#include <hip/hip_runtime.h>

typedef float v2f __attribute__((ext_vector_type(2)));
typedef float v8f __attribute__((ext_vector_type(8)));

#define WPB 8  // waves per block (256 threads, wave32)

// ---------------------------------------------------------------------------
// Kernel 0: zero the scatter accumulators in workspace
// ---------------------------------------------------------------------------
__global__ void zero_f32(float* __restrict__ p, int n) {
  int i = blockIdx.x * blockDim.x + threadIdx.x;
  if (i < n) p[i] = 0.0f;
}

// ---------------------------------------------------------------------------
// Kernel 1: edge scatter — segment sums + counts via native f32 atomics.
// Index stream (256MB) is the HBM cost; targets are L2-resident (192MB L2).
// ---------------------------------------------------------------------------
__global__ __launch_bounds__(256) void edge_scatter(
    const int* __restrict__ src, const int* __restrict__ dst,
    const float* __restrict__ nf, float* __restrict__ sums,
    float* __restrict__ cnt, int n_edges) {
  int e = blockIdx.x * blockDim.x + threadIdx.x;
  if (e >= n_edges) return;
  int s = src[e];
  int d = dst[e];
  float f0 = nf[2 * s + 0];
  float f1 = nf[2 * s + 1];
  unsafeAtomicAdd(&sums[2 * d + 0], f0);   // global_atomic_add_f32
  unsafeAtomicAdd(&sums[2 * d + 1], f1);
  unsafeAtomicAdd(&cnt[d], 1.0f);
}

__device__ __forceinline__ float sigmoid_fast(float x) {
  return 1.0f / (1.0f + __expf(-x));
}
__device__ __forceinline__ float tanh_fast(float x) {
  return 2.0f * sigmoid_fast(2.0f * x) - 1.0f;
}

// ---------------------------------------------------------------------------
// Kernel 2: per-node dense chain with V_WMMA_F32_16X16X4_F32.
// One wave processes a tile of 16 nodes.
//   Stage 1: x[16,16] = agg[16,2 (pad K=4)] @ W_msg^T + b_msg   (1 WMMA)
//   Stage 2: gi[16,48] = x[16,16] @ W_ih^T + b_ih               (3x4 WMMAs)
//   Stage 3: GRU gates with h0=0  (gh == b_hh, W_hh drops out)
//   Stage 4: out[16,2] = h @ W_cls^T + b_cls (scalar via LDS)
// f32 WMMA layouts (ISA 7.12.2):
//   A 16x4 : lanes 0-15 row M=lane  {v0=K0, v1=K1}; lanes 16-31 {v0=K2, v1=K3}
//   B 4x16 : lanes 0-15 col N=lane  {v0=K0, v1=K1}; lanes 16-31 {v0=K2, v1=K3}
//   C/D    : vgpr i: lanes 0-15 -> (M=i, N=lane), lanes 16-31 -> (M=i+8, N=lane-16)
// ---------------------------------------------------------------------------
__global__ __launch_bounds__(256) void node_update_wmma(
    const float* __restrict__ nf, const float* __restrict__ sums,
    const float* __restrict__ cnt,
    const float* __restrict__ W_msg, const float* __restrict__ b_msg,
    const float* __restrict__ W_ih,  const float* __restrict__ b_ih,
    const float* __restrict__ b_hh,
    const float* __restrict__ W_cls, const float* __restrict__ b_cls,
    float* __restrict__ out, int n_tiles) {
  __shared__ float lds[WPB][16][17];  // per-wave 16x16 staging tile (padded)

  const int lane = threadIdx.x & 31;
  const int wave = threadIdx.x >> 5;
  const int tile = blockIdx.x * WPB + wave;
  if (tile >= n_tiles) return;

  const int half = lane >> 4;   // 0: lanes 0-15, 1: lanes 16-31
  const int l = lane & 15;

  // ---- A: agg tile in 16x4 f32 A-layout (K=0,1 live; K=2,3 zero pad) ----
  v2f a;
  if (half == 0) {
    int node = tile * 16 + l;
    float c = cnt[node];
    float f0, f1;
    if (c > 0.0f) {
      float inv = 1.0f / c;
      f0 = sums[2 * node + 0] * inv;
      f1 = sums[2 * node + 1] * inv;
    } else {
      f0 = nf[2 * node + 0];
      f1 = nf[2 * node + 1];
    }
    a.x = f0; a.y = f1;
  } else {
    a.x = 0.0f; a.y = 0.0f;
  }

  // ---- B: W_msg^T padded to 4x16.  B[k][n] = W_msg[n][k] ----
  v2f b;
  if (half == 0) { b.x = W_msg[l * 2 + 0]; b.y = W_msg[l * 2 + 1]; }
  else           { b.x = 0.0f;             b.y = 0.0f; }

  // ---- C preloaded with b_msg broadcast down columns ----
  v8f c;
  {
    float bm = b_msg[l];
#pragma unroll
    for (int i = 0; i < 8; ++i) c[i] = bm;
  }

  v8f x = __builtin_amdgcn_wmma_f32_16x16x4_f32(
      false, a, false, b, (short)0, c, false, false);

  // ---- stage x through per-wave LDS to rebuild it in A-layout ----
  float (*X)[17] = lds[wave];
#pragma unroll
  for (int i = 0; i < 8; ++i) X[i + 8 * half][l] = x[i];
  asm volatile("s_wait_dscnt 0" ::: "memory");

  v2f xa[4];
#pragma unroll
  for (int kc = 0; kc < 4; ++kc) {
    int k0 = 4 * kc + 2 * half;          // row M = l in both halves
    xa[kc].x = X[l][k0 + 0];
    xa[kc].y = X[l][k0 + 1];
  }
  asm volatile("s_wait_dscnt 0" ::: "memory");

  // ---- gi tiles: r (cols 0-15), z (16-31), n (32-47) ----
  v8f gi[3];
#pragma unroll
  for (int t = 0; t < 3; ++t) {
    v8f acc;
    float bi = b_ih[t * 16 + l];
#pragma unroll
    for (int i = 0; i < 8; ++i) acc[i] = bi;
#pragma unroll
    for (int kc = 0; kc < 4; ++kc) {
      // B[k][n] = W_ih^T[k][t*16+n] = W_ih[t*16+n][k]
      int row = t * 16 + l;
      int k = 4 * kc + 2 * half;
      v2f bb;
      bb.x = W_ih[row * 16 + k + 0];
      bb.y = W_ih[row * 16 + k + 1];
      acc = __builtin_amdgcn_wmma_f32_16x16x4_f32(
          false, xa[kc], false, bb, (short)0, acc, false, false);
    }
    gi[t] = acc;
  }

  // ---- GRU gates, h0 = 0 => gh = b_hh; h = (1-z)*n ----
  float bhr = b_hh[0 * 16 + l];
  float bhz = b_hh[1 * 16 + l];
  float bhn = b_hh[2 * 16 + l];
  v8f h;
#pragma unroll
  for (int i = 0; i < 8; ++i) {
    float r  = sigmoid_fast(gi[0][i] + bhr);
    float z  = sigmoid_fast(gi[1][i] + bhz);
    float nn = tanh_fast(gi[2][i] + r * bhn);
    h[i] = (1.0f - z) * nn;
  }

  // ---- classifier: stage h to LDS, one (node, class) pair per lane ----
#pragma unroll
  for (int i = 0; i < 8; ++i) X[i + 8 * half][l] = h[i];
  asm volatile("s_wait_dscnt 0" ::: "memory");

  {
    int m  = lane >> 1;       // local node 0..15
    int cc = lane & 1;        // class 0..1
    float acc = b_cls[cc];
#pragma unroll
    for (int k = 0; k < 16; ++k) acc += X[m][k] * W_cls[cc * 16 + k];
    int node = tile * 16 + m;
    out[node * 2 + cc] = acc;
  }
}

// ---------------------------------------------------------------------------
// Scalar fallback for a tail when n_nodes % 16 != 0 (WMMA needs full tiles).
// ---------------------------------------------------------------------------
__global__ void node_update_scalar(
    const float* __restrict__ nf, const float* __restrict__ sums,
    const float* __restrict__ cnt,
    const float* __restrict__ W_msg, const float* __restrict__ b_msg,
    const float* __restrict__ W_ih,  const float* __restrict__ b_ih,
    const float* __restrict__ b_hh,
    const float* __restrict__ W_cls, const float* __restrict__ b_cls,
    float* __restrict__ out, int start, int n_nodes) {
  int node = start + blockIdx.x * blockDim.x + threadIdx.x;
  if (node >= n_nodes) return;
  float c = cnt[node];
  float a0, a1;
  if (c > 0.0f) { a0 = sums[2 * node] / c; a1 = sums[2 * node + 1] / c; }
  else          { a0 = nf[2 * node];       a1 = nf[2 * node + 1]; }
  float x[16];
#pragma unroll
  for (int k = 0; k < 16; ++k)
    x[k] = W_msg[2 * k] * a0 + W_msg[2 * k + 1] * a1 + b_msg[k];
  float h[16];
#pragma unroll
  for (int j = 0; j < 16; ++j) {
    float gr = b_ih[j], gz = b_ih[16 + j], gn = b_ih[32 + j];
#pragma unroll
    for (int k = 0; k < 16; ++k) {
      gr += x[k] * W_ih[(j)      * 16 + k];
      gz += x[k] * W_ih[(16 + j) * 16 + k];
      gn += x[k] * W_ih[(32 + j) * 16 + k];
    }
    float r  = sigmoid_fast(gr + b_hh[j]);
    float z  = sigmoid_fast(gz + b_hh[16 + j]);
    float nn = tanh_fast(gn + r * b_hh[32 + j]);
    h[j] = (1.0f - z) * nn;
  }
#pragma unroll
  for (int cc = 0; cc < 2; ++cc) {
    float acc = b_cls[cc];
#pragma unroll
    for (int k = 0; k < 16; ++k) acc += h[k] * W_cls[cc * 16 + k];
    out[node * 2 + cc] = acc;
  }
}

// ---------------------------------------------------------------------------
extern "C" void kernel_launch(void* const* d_in, const int* in_sizes, int n_in,
                              void* d_out, int out_size, void* d_ws,
                              size_t ws_size, hipStream_t stream) {
  const float* nf    = (const float*)d_in[0];   // [N,2]
  const int*   ei    = (const int*)  d_in[1];   // [2,E]
  const float* W_msg = (const float*)d_in[2];   // [16,2]
  const float* b_msg = (const float*)d_in[3];   // [16]
  const float* W_ih  = (const float*)d_in[4];   // [48,16]
  // d_in[5] = W_hh: unused (h0 == 0)
  const float* b_ih  = (const float*)d_in[6];   // [48]
  const float* b_hh  = (const float*)d_in[7];   // [48]
  const float* W_cls = (const float*)d_in[8];   // [2,16]
  const float* b_cls = (const float*)d_in[9];   // [2]
  float* out = (float*)d_out;

  const int n_nodes = in_sizes[0] / 2;
  const int n_edges = in_sizes[1] / 2;
  const int* src = ei;
  const int* dst = ei + n_edges;

  float* sums = (float*)d_ws;                 // [N,2]
  float* cnt  = sums + (size_t)2 * n_nodes;   // [N]

  // zero accumulators (harness poisons d_ws; must re-zero every call)
  int tot = 3 * n_nodes;
  zero_f32<<<(tot + 255) / 256, 256, 0, stream>>>(sums, tot);

  // edge scatter
  edge_scatter<<<(n_edges + 255) / 256, 256, 0, stream>>>(
      src, dst, nf, sums, cnt, n_edges);

  // per-node WMMA chain, 16 nodes per wave
  int n_tiles = n_nodes / 16;
  if (n_tiles > 0) {
    int blocks = (n_tiles + WPB - 1) / WPB;
    node_update_wmma<<<blocks, 32 * WPB, 0, stream>>>(
        nf, sums, cnt, W_msg, b_msg, W_ih, b_ih, b_hh, W_cls, b_cls, out,
        n_tiles);
  }

  // scalar tail (N=1M is a multiple of 16, so this normally doesn't launch)
  int rem = n_nodes - n_tiles * 16;
  if (rem > 0) {
    node_update_scalar<<<(rem + 63) / 64, 64, 0, stream>>>(
        nf, sums, cnt, W_msg, b_msg, W_ih, b_ih, b_hh, W_cls, b_cls, out,
        n_tiles * 16, n_nodes);
  }
}